// EdgeAggregation_57930518888711
// MI455X (gfx1250) — compile-verified
//
#include <hip/hip_runtime.h>
#include <math.h>

// ---------------------------------------------------------------------------
// Problem constants (from the reference):
//   receivers: [8, 2048, 128] f32   (d_in[0])
//   senders:   [8, 2048, 128] f32   (d_in[1])
//   out: edges [8*2048*16, 256] f32  followed by  sender [8, 2048, 2048] f32
// ---------------------------------------------------------------------------
#define BATCH 8
#define NPTS  2048
#define FDIM  128
#define KNN   16

typedef __attribute__((ext_vector_type(2))) float v2f;
typedef __attribute__((ext_vector_type(8))) float v8f;

// ---------------------------------------------------------------------------
// Phase A: fused distance (WMMA f32 16x16x4) + running top-16 per sender.
// One wave (32 lanes) per block handles 16 senders; loops over 128 receiver
// tiles. Distances never leave the WGP; only 16 indices/sender hit memory.
// ---------------------------------------------------------------------------
__global__ __launch_bounds__(32)
void knn_topk_wmma(const float* __restrict__ senders,
                   const float* __restrict__ receivers,
                   int* __restrict__ topk) {
  const int tile = blockIdx.x;       // sender tile: 0..127
  const int b    = blockIdx.y;       // batch
  const int lane = threadIdx.x;      // 0..31 (wave32)
  const int half = lane >> 4;        // 0 or 1
  const int row  = lane & 15;

  __shared__ float dtile[16 * 16];

  // ---- preload A (senders) fragments: lane l holds S[l%16][4t + 2*half + j]
  const float* Sb =
      senders + ((size_t)b * NPTS + (size_t)tile * 16 + row) * FDIM + 2 * half;
  v2f sfrag[32];
  float x2p = 0.f;
#pragma unroll
  for (int t = 0; t < 32; ++t) {
    sfrag[t] = *(const v2f*)(Sb + 4 * t);
    x2p = fmaf(sfrag[t].x, sfrag[t].x, x2p);
    x2p = fmaf(sfrag[t].y, sfrag[t].y, x2p);
  }
  // complete ||s||^2 across the two lane-halves (each half saw half of K)
  float x2full = x2p + __shfl_xor(x2p, 16);
  // C-layout row bias: VGPR v of lane l corresponds to sender row v + 8*half
  float x2bc[8];
#pragma unroll
  for (int v = 0; v < 8; ++v) x2bc[v] = __shfl(x2full, v + 8 * half);

  // ---- per-sender top-16 (sorted ascending by distance), lanes 0..15 own rows
  float best_d[KNN];
  int   best_i[KNN];
#pragma unroll
  for (int j = 0; j < KNN; ++j) { best_d[j] = __builtin_inff(); best_i[j] = 0; }

  const float* Rbase =
      receivers + (size_t)b * NPTS * FDIM + (size_t)row * FDIM + 2 * half;

  for (int rt = 0; rt < NPTS / 16; ++rt) {
    const float* Rb = Rbase + (size_t)rt * 16 * FDIM;
    if (rt + 1 < NPTS / 16)
      __builtin_prefetch(Rb + 16 * FDIM, 0, 3);   // WGP-scope global_prefetch

    // B (receivers^T) fragments: identical per-lane addressing as A
    v2f rfrag[32];
    float y2p = 0.f;
#pragma unroll
    for (int t = 0; t < 32; ++t) {
      rfrag[t] = *(const v2f*)(Rb + 4 * t);
      y2p = fmaf(rfrag[t].x, rfrag[t].x, y2p);
      y2p = fmaf(rfrag[t].y, rfrag[t].y, y2p);
    }
    float y2full = y2p + __shfl_xor(y2p, 16);   // ||r_n||^2 for n = lane%16

    // 16x16 tile of S . R^T, K=128 via 32 WMMAs in two independent chains
    v8f acc0 = {};
    v8f acc1 = {};
#pragma unroll
    for (int t = 0; t < 16; ++t) {
      acc0 = __builtin_amdgcn_wmma_f32_16x16x4_f32(
          false, sfrag[2 * t],     false, rfrag[2 * t],
          (short)0, acc0, false, false);
      acc1 = __builtin_amdgcn_wmma_f32_16x16x4_f32(
          false, sfrag[2 * t + 1], false, rfrag[2 * t + 1],
          (short)0, acc1, false, false);
    }

    // dist[m][n] = |x2[m] + y2[n] - 2*mixed|, stage through LDS for top-k
#pragma unroll
    for (int v = 0; v < 8; ++v) {
      float mixed = acc0[v] + acc1[v];
      float d = fabsf(fmaf(-2.f, mixed, x2bc[v] + y2full));
      dtile[(v + 8 * half) * 16 + row] = d;
    }
    __syncthreads();

    if (lane < 16) {
      const int ibase = rt * 16;
#pragma unroll
      for (int n = 0; n < 16; ++n) {
        float d = dtile[lane * 16 + n];
        if (d < best_d[KNN - 1]) {          // beats current 16th best
          float dd = d;
          int   ii = ibase + n;
#pragma unroll
          for (int j = 0; j < KNN; ++j) {   // register bubble-insert (sorted)
            bool  sw = dd < best_d[j];
            float td = best_d[j];
            int   ti = best_i[j];
            best_d[j] = sw ? dd : td;
            best_i[j] = sw ? ii : ti;
            dd = sw ? td : dd;
            ii = sw ? ti : ii;
          }
        }
      }
    }
    __syncthreads();
  }

  if (lane < 16) {
    // sort indices ascending (reference does jnp.sort(idx)) — odd-even network
#pragma unroll
    for (int p = 0; p < KNN; ++p) {
#pragma unroll
      for (int j = (p & 1); j < KNN - 1; j += 2) {
        if (best_i[j] > best_i[j + 1]) {
          int tmp = best_i[j];
          best_i[j] = best_i[j + 1];
          best_i[j + 1] = tmp;
        }
      }
    }
    int* dst = topk + ((size_t)b * NPTS + (size_t)tile * 16 + lane) * KNN;
#pragma unroll
    for (int j = 0; j < KNN; ++j) dst[j] = best_i[j];
  }
}

// ---------------------------------------------------------------------------
// Phase B: pure-bandwidth output materialization.
// One block per (b, s): writes 16 edge rows (256 f32 each) and the 2048-f32
// one-hot connectivity row. All float4, fully coalesced.
// ---------------------------------------------------------------------------
__global__ __launch_bounds__(256)
void emit_outputs(const float* __restrict__ senders,
                  const float* __restrict__ receivers,
                  const int* __restrict__ topk,
                  float* __restrict__ out) {
  const int bs = blockIdx.x;          // b*NPTS + s
  const int b  = bs >> 11;            // / 2048
  const int t  = threadIdx.x;

  const int* idx = topk + (size_t)bs * KNN;

  float* edges = out;                                         // [B*N*K, 256]
  float* smat  = out + (size_t)BATCH * NPTS * KNN * (2 * FDIM);

  const float4* sf4 = (const float4*)(senders + (size_t)bs * FDIM);
  const float4* rb4 = (const float4*)(receivers + (size_t)b * NPTS * FDIM);
  float4* e4 = (float4*)edges + (size_t)bs * KNN * 64;        // 64 f4 per row

  const int c4 = t & 63;              // float4 column within the 256-f32 row
  const int jg = t >> 6;              // edge-row group (4 rows in flight)

  float4 sval;
  if (c4 < 32) sval = sf4[c4];        // sender half, broadcast to all 16 rows

#pragma unroll
  for (int jj = 0; jj < 4; ++jj) {
    const int j = jg + 4 * jj;        // edge row 0..15
    float4 v;
    if (c4 < 32) {
      v = sval;
    } else {
      const int r = idx[j];
      v = rb4[(size_t)r * (FDIM / 4) + (c4 - 32)];
    }
    e4[(size_t)j * 64 + c4] = v;
  }

  // one-hot connectivity row: zero-fill then scatter 16 ones
  float4* sm4 = (float4*)(smat + (size_t)bs * NPTS);
  float4 z = {0.f, 0.f, 0.f, 0.f};
  sm4[t] = z;
  sm4[t + 256] = z;
  __syncthreads();
  if (t < KNN) smat[(size_t)bs * NPTS + idx[t]] = 1.0f;
}

// ---------------------------------------------------------------------------
extern "C" void kernel_launch(void* const* d_in, const int* in_sizes, int n_in,
                              void* d_out, int out_size, void* d_ws,
                              size_t ws_size, hipStream_t stream) {
  const float* receivers = (const float*)d_in[0];
  const float* senders   = (const float*)d_in[1];
  float* out = (float*)d_out;
  int* topk = (int*)d_ws;             // B*NPTS*KNN ints = 1 MB scratch

  dim3 gridA(NPTS / 16, BATCH);       // 128 x 8 = 1024 waves
  knn_topk_wmma<<<gridA, 32, 0, stream>>>(senders, receivers, topk);

  emit_outputs<<<BATCH * NPTS, 256, 0, stream>>>(senders, receivers, topk,
                                                 out);
}